// MF_84722524880963
// MI455X (gfx1250) — compile-verified
//
#include <hip/hip_runtime.h>

// CDNA5 / gfx1250, wave32. Embedding gather + batched dot via V_WMMA_F32_16X16X4_F32.
// Each wave handles 16 batch rows: D = U(16x64) * V^T(64x16) accumulated in
// 16 chained 16x16x4 fp32 WMMAs; diag(D) is the per-row dot product.

typedef __attribute__((ext_vector_type(2))) float v2f;
typedef __attribute__((ext_vector_type(8))) float v8f;

#define EMBED_DIM 64
#define ITEM_OFFSET 100000   // cumsum offset of field 1 (FIELD_DIMS=[100000,1000000])
#define WAVES_PER_BLOCK 8
#define ROWS_PER_WAVE 16

__global__ __launch_bounds__(256) void MF_84722524880963_kernel(
    const int* __restrict__ x,       // (B, 2) int32 raw field indices
    const float* __restrict__ table, // (1.1M, 64) fp32 embedding table
    float* __restrict__ out,         // (B,) fp32 dot products
    int nrows)
{
  const int lane = threadIdx.x & 31;
  const int wave = threadIdx.x >> 5;
  const int rowbase = (blockIdx.x * WAVES_PER_BLOCK + wave) * ROWS_PER_WAVE;

  const int mn = lane & 15;   // M (A half) / N (B half) index for this lane
  const int hi = lane >> 4;   // 0: lanes 0-15 (K,K+1), 1: lanes 16-31 (K+2,K+3)

  // Clamp so every lane issues valid loads (EXEC must stay all-1s for WMMA).
  int row = rowbase + mn;
  if (row >= nrows) row = nrows - 1;

  // Gather table row pointers. user field -> offset 0, item field -> +100000.
  const int iu = x[2 * row + 0];
  const int iv = x[2 * row + 1] + ITEM_OFFSET;
  const float* __restrict__ urow = table + (size_t)iu * EMBED_DIM;
  const float* __restrict__ vrow = table + (size_t)iv * EMBED_DIM;

  // Two accumulator chains to relax the WMMA C->C RAW dependency.
  v8f acc0 = {};
  v8f acc1 = {};

#pragma unroll
  for (int k0 = 0; k0 < EMBED_DIM; k0 += 8) {
    const int ka = k0 + 2 * hi;
    // A tile (user rows) and B tile (item rows, fed transposed by layout):
    // float2 -> VGPR pair {K, K+1} exactly matching the 16x16x4 f32 layout.
    v2f a0 = *(const v2f*)(urow + ka);
    v2f b0 = *(const v2f*)(vrow + ka);
    v2f a1 = *(const v2f*)(urow + ka + 4);
    v2f b1 = *(const v2f*)(vrow + ka + 4);

    acc0 = __builtin_amdgcn_wmma_f32_16x16x4_f32(
        /*neg_a=*/false, a0, /*neg_b=*/false, b0,
        /*c_mod=*/(short)0, acc0, /*reuse_a=*/false, /*reuse_b=*/false);
    acc1 = __builtin_amdgcn_wmma_f32_16x16x4_f32(
        /*neg_a=*/false, a1, /*neg_b=*/false, b1,
        /*c_mod=*/(short)0, acc1, /*reuse_a=*/false, /*reuse_b=*/false);
  }

  v8f acc = acc0 + acc1;

  // Diagonal extraction from the 16x16 f32 C/D layout:
  //   m in 0..7  -> lane m,    component m
  //   m in 8..15 -> lane 16+m, component m-8   (i.e. lane 24+r holds comp r)
  int r = -1;
  int mdiag = 0;
  if (lane < 8)        { r = lane;      mdiag = r;     }
  else if (lane >= 24) { r = lane - 24; mdiag = r + 8; }

  if (r >= 0) {
    float d = acc[0];
    d = (r == 1) ? acc[1] : d;
    d = (r == 2) ? acc[2] : d;
    d = (r == 3) ? acc[3] : d;
    d = (r == 4) ? acc[4] : d;
    d = (r == 5) ? acc[5] : d;
    d = (r == 6) ? acc[6] : d;
    d = (r == 7) ? acc[7] : d;
    const int orow = rowbase + mdiag;
    if (orow < nrows) out[orow] = d;
  }
}

extern "C" void kernel_launch(void* const* d_in, const int* in_sizes, int n_in,
                              void* d_out, int out_size, void* d_ws, size_t ws_size,
                              hipStream_t stream) {
  const int*   x     = (const int*)d_in[0];    // (B,2) int32
  const float* table = (const float*)d_in[1];  // (1.1M,64) fp32
  float*       out   = (float*)d_out;          // (B,) fp32

  const int nrows = in_sizes[0] / 2;           // B = 16384
  const int rows_per_block = ROWS_PER_WAVE * WAVES_PER_BLOCK;  // 128
  const int blocks = (nrows + rows_per_block - 1) / rows_per_block;

  MF_84722524880963_kernel<<<blocks, 256, 0, stream>>>(x, table, out, nrows);
}